// GraphEncoder_43748536877364
// MI455X (gfx1250) — compile-verified
//
#include <hip/hip_runtime.h>
#include <hip/hip_bf16.h>

#define NN 100000
#define BB 256
#define HH 128
#define EE 600000
#define LL 2

typedef float v2f __attribute__((ext_vector_type(2)));
typedef float v8f __attribute__((ext_vector_type(8)));

// ---------------- utility ----------------
__global__ void fill_zero(float* p, long n) {
    long i = (long)blockIdx.x * blockDim.x + threadIdx.x;
    long stride = (long)gridDim.x * blockDim.x;
    for (; i < n; i += stride) p[i] = 0.f;
}

// ---------------- degrees / norms ----------------
__global__ void degree_kernel(const int* __restrict__ edges,
                              float* __restrict__ deg_src,
                              float* __restrict__ deg_dst) {
    int idx = blockIdx.x * blockDim.x + threadIdx.x;
    if (idx >= 4 * EE) return;
    int r = idx / EE, e = idx - r * EE;
    int s = edges[(size_t)r * 2 * EE + e];
    int d = edges[(size_t)r * 2 * EE + EE + e];
    atomicAdd(&deg_src[(size_t)r * NN + s], 1.f);
    atomicAdd(&deg_dst[(size_t)r * NN + d], 1.f);
}

__global__ void rsqrt_kernel(float* p, int n) {
    int i = blockIdx.x * blockDim.x + threadIdx.x;
    if (i < n) p[i] = rsqrtf(fmaxf(p[i], 1.f));
}

// ---------------- feature init ----------------
__global__ void init_patient(const float* __restrict__ emb, float* __restrict__ h) {
    int i = blockIdx.x * blockDim.x + threadIdx.x;
    if (i < NN * HH) h[i] = emb[i & (HH - 1)];
}

// one wave32 per node: 32 lanes x float4 = 128 features; layernorm via shfl reductions
__global__ void init_ln(const float* __restrict__ table, const int* __restrict__ tid,
                        const float* __restrict__ g, const float* __restrict__ b,
                        float* __restrict__ out) {
    int wave = (blockIdx.x * blockDim.x + threadIdx.x) >> 5;
    int lane = threadIdx.x & 31;
    if (wave >= NN) return;
    int t = tid[wave];
    float4 x = ((const float4*)(table + (size_t)t * HH))[lane];
    float s = x.x + x.y + x.z + x.w;
#pragma unroll
    for (int m = 16; m >= 1; m >>= 1) s += __shfl_xor(s, m, 32);
    float mu = s * (1.f / HH);
    float dx = x.x - mu, dy = x.y - mu, dz = x.z - mu, dw = x.w - mu;
    float ss = dx * dx + dy * dy + dz * dz + dw * dw;
#pragma unroll
    for (int m = 16; m >= 1; m >>= 1) ss += __shfl_xor(ss, m, 32);
    float inv = rsqrtf(ss * (1.f / HH) + 1e-5f);
    float4 g4 = ((const float4*)g)[lane];
    float4 b4 = ((const float4*)b)[lane];
    float4 o;
    o.x = dx * inv * g4.x + b4.x;
    o.y = dy * inv * g4.y + b4.y;
    o.z = dz * inv * g4.z + b4.z;
    o.w = dw * inv * g4.w + b4.w;
    ((float4*)(out + (size_t)wave * HH))[lane] = o;
}

// ---------------- message scatter (gather h[src]*norm, atomic-add to agg[dst]) ----------------
__global__ void scatter_msg(const float* __restrict__ hsrc, const float* __restrict__ nsrc,
                            const int* __restrict__ src, const int* __restrict__ dst,
                            float* __restrict__ agg) {
    int wave = (blockIdx.x * blockDim.x + threadIdx.x) >> 5;
    int lane = threadIdx.x & 31;
    if (wave >= EE) return;
    int s = src[wave], d = dst[wave];
    float ns = nsrc[s];
    float4 m = ((const float4*)(hsrc + (size_t)s * HH))[lane];
    float* ap = agg + (size_t)d * HH + lane * 4;
    atomicAdd(ap + 0, m.x * ns);
    atomicAdd(ap + 1, m.y * ns);
    atomicAdd(ap + 2, m.z * ns);
    atomicAdd(ap + 3, m.w * ns);
}

// ---------------- f32 WMMA GEMM (row-strip variant, Nout == HH == 128) ----------------
// One wave computes a 16x128 output strip: A (16x4) loaded once per K-step, fed to 8
// WMMAs with distinct accumulators (no back-to-back same-D hazards).
// out[16x128] (+)= (A_strip @ B) * rowscale + bias
template <bool ACC>
__global__ void wmma_gemm_rowstrip(const float* __restrict__ A, const float* __restrict__ Bm,
                                   const float* __restrict__ bias,
                                   const float* __restrict__ rowscale,
                                   float* __restrict__ out, int M) {
    int waveId = (blockIdx.x * blockDim.x + threadIdx.x) >> 5;
    int lane = threadIdx.x & 31;
    int numMt = M >> 4;
    if (waveId >= numMt) return;      // wave-uniform: EXEC all-ones at WMMA
    int mloc = lane & 15;             // M within A-tile / N within B,C,D tile
    int kh = lane >> 4;               // K-half selector
    const float* arow = A + (size_t)(waveId * 16 + mloc) * HH + 2 * kh;
    const float* bbase = Bm + (size_t)(2 * kh) * HH + mloc;
    v8f c[8];
#pragma unroll
    for (int t = 0; t < 8; ++t) c[t] = (v8f){0.f, 0.f, 0.f, 0.f, 0.f, 0.f, 0.f, 0.f};
    for (int k = 0; k < HH; k += 4) {
        v2f a;
        a[0] = arow[k];
        a[1] = arow[k + 1];
        const float* bp = bbase + (size_t)k * HH;
#pragma unroll
        for (int t = 0; t < 8; ++t) {
            v2f bv;
            bv[0] = bp[t * 16];
            bv[1] = bp[t * 16 + HH];
            c[t] = __builtin_amdgcn_wmma_f32_16x16x4_f32(false, a, false, bv,
                                                         (short)0, c[t], false, false);
        }
    }
    float bcol[8];
#pragma unroll
    for (int t = 0; t < 8; ++t) bcol[t] = bias[t * 16 + mloc];
#pragma unroll
    for (int v = 0; v < 8; ++v) {
        int row = waveId * 16 + v + 8 * kh;
        float rs = rowscale[row];
        float* orow = out + (size_t)row * HH;
#pragma unroll
        for (int t = 0; t < 8; ++t) {
            float val = c[t][v] * rs + bcol[t];
            int col = t * 16 + mloc;
            if (ACC) orow[col] += val; else orow[col] = val;
        }
    }
}

// ---------------- generic per-tile f32 WMMA GEMM (used for the 256x128x512 projection) ----
__global__ void wmma_gemm_tile(const float* __restrict__ A, const float* __restrict__ Bm,
                               const float* __restrict__ bias, float* __restrict__ out,
                               int M, int Nout) {
    int waveId = (blockIdx.x * blockDim.x + threadIdx.x) >> 5;
    int lane = threadIdx.x & 31;
    int numNt = Nout >> 4;
    int numMt = M >> 4;
    int mt = waveId / numNt;
    int nt = waveId - mt * numNt;
    if (mt >= numMt) return;
    int mloc = lane & 15;
    int kh = lane >> 4;
    const float* arow = A + (size_t)(mt * 16 + mloc) * HH + 2 * kh;
    const float* bcol = Bm + (size_t)(2 * kh) * Nout + nt * 16 + mloc;
    v8f c = {0.f, 0.f, 0.f, 0.f, 0.f, 0.f, 0.f, 0.f};
#pragma unroll 4
    for (int k = 0; k < HH; k += 4) {
        v2f a, bv;
        a[0] = arow[k];
        a[1] = arow[k + 1];
        bv[0] = bcol[(size_t)k * Nout];
        bv[1] = bcol[(size_t)(k + 1) * Nout];
        c = __builtin_amdgcn_wmma_f32_16x16x4_f32(false, a, false, bv,
                                                  (short)0, c, false, false);
    }
    int col = nt * 16 + mloc;
    float bb = bias[col];
#pragma unroll
    for (int v = 0; v < 8; ++v) {
        int row = mt * 16 + v + 8 * kh;
        out[(size_t)row * Nout + col] = c[v] + bb;
    }
}

__global__ void relu_kernel(float* p, long n) {
    long i = (long)blockIdx.x * blockDim.x + threadIdx.x;
    long stride = (long)gridDim.x * blockDim.x;
    for (; i < n; i += stride) p[i] = fmaxf(p[i], 0.f);
}

// ---------------- pooling ----------------
__global__ void pool_accum(const float* __restrict__ h, const int* __restrict__ gid,
                           float* __restrict__ sums, float* __restrict__ cnt) {
    int wave = (blockIdx.x * blockDim.x + threadIdx.x) >> 5;
    int lane = threadIdx.x & 31;
    if (wave >= NN) return;
    int g = gid[wave];
    float4 v = ((const float4*)(h + (size_t)wave * HH))[lane];
    float* sp = sums + (size_t)g * HH + lane * 4;
    atomicAdd(sp + 0, v.x);
    atomicAdd(sp + 1, v.y);
    atomicAdd(sp + 2, v.z);
    atomicAdd(sp + 3, v.w);
    if (lane == 0) atomicAdd(&cnt[g], 1.f);
}

__global__ void pool_combine(const float* __restrict__ sums, const float* __restrict__ cnt,
                             float* __restrict__ pooled) {
    int i = blockIdx.x * blockDim.x + threadIdx.x;
    if (i >= BB * HH) return;
    float cv = cnt[i >> 7];
    pooled[i] += (cv > 0.f) ? sums[i] / cv : 0.f;
}

// ---------------- final L2 normalize over 512-wide rows ----------------
__global__ void l2norm_kernel(float* __restrict__ out) {
    __shared__ float red[256];
    int row = blockIdx.x, t = threadIdx.x;
    float a = out[(size_t)row * 512 + t];
    float b = out[(size_t)row * 512 + 256 + t];
    red[t] = a * a + b * b;
    __syncthreads();
#pragma unroll
    for (int s = 128; s >= 1; s >>= 1) {
        if (t < s) red[t] += red[t + s];
        __syncthreads();
    }
    float denom = fmaxf(sqrtf(red[0]), 1e-6f);
    out[(size_t)row * 512 + t] = a / denom;
    out[(size_t)row * 512 + 256 + t] = b / denom;
}

// ---------------- host orchestration ----------------
extern "C" void kernel_launch(void* const* d_in, const int* in_sizes, int n_in,
                              void* d_out, int out_size, void* d_ws, size_t ws_size,
                              hipStream_t stream) {
    const float* diag_table = (const float*)d_in[0];
    const float* desc_table = (const float*)d_in[1];
    const float* ln_g   = (const float*)d_in[2];
    const float* ln_b   = (const float*)d_in[3];
    const float* emb_pat = (const float*)d_in[4];
    const float* conv_W = (const float*)d_in[5];
    const float* conv_b = (const float*)d_in[6];
    const float* proj_W = (const float*)d_in[7];
    const float* proj_b = (const float*)d_in[8];
    const int* tid_diag = (const int*)d_in[9];
    const int* tid_desc = (const int*)d_in[10];
    const int* edges    = (const int*)d_in[11];
    const int* gids[3]  = {(const int*)d_in[12], (const int*)d_in[13], (const int*)d_in[14]};

    const size_t NH = (size_t)NN * HH;
    float* ws    = (float*)d_ws;
    float* hbuf  = ws;                       // 3*NH
    float* obuf  = hbuf + 3 * NH;            // 3*NH
    float* agg   = obuf + 3 * NH;            // NH
    float* norms = agg + NH;                 // 8*NN: [0,4N)=src norms, [4N,8N)=dst norms
    float* sums  = norms + 8 * (size_t)NN;   // BB*HH
    float* cnt   = sums + (size_t)BB * HH;   // BB (contiguous after sums)
    float* pooled = cnt + BB;                // BB*HH

    auto gs = [](long n, int b) { return (int)((n + b - 1) / b); };

    // degree norms (symmetric, per relation)
    fill_zero<<<1024, 256, 0, stream>>>(norms, 8L * NN);
    degree_kernel<<<gs(4L * EE, 256), 256, 0, stream>>>(edges, norms, norms + 4L * NN);
    rsqrt_kernel<<<gs(8L * NN, 256), 256, 0, stream>>>(norms, 8 * NN);

    // initial features
    init_patient<<<gs((long)NN * HH, 256), 256, 0, stream>>>(emb_pat, hbuf);
    init_ln<<<gs(NN, 8), 256, 0, stream>>>(diag_table, tid_diag, ln_g, ln_b, hbuf + NH);
    init_ln<<<gs(NN, 8), 256, 0, stream>>>(desc_table, tid_desc, ln_g, ln_b, hbuf + 2 * NH);

    const int ST[4] = {1, 2, 0, 0};
    const int DT[4] = {0, 0, 1, 2};

    float* cur = hbuf;
    float* nxt = obuf;
    const long strip_threads = (long)(NN / 16) * 32;  // one wave per 16x128 row strip
    for (int layer = 0; layer < LL; ++layer) {
        for (int r = 0; r < 4; ++r) {
            fill_zero<<<1024, 256, 0, stream>>>(agg, (long)NH);
            const int* src = edges + (size_t)r * 2 * EE;
            const int* dst = src + EE;
            scatter_msg<<<gs(EE, 8), 256, 0, stream>>>(cur + (size_t)ST[r] * NH,
                                                       norms + (size_t)r * NN, src, dst, agg);
            const float* Wl = conv_W + ((size_t)layer * 4 + r) * HH * HH;
            const float* bl = conv_b + ((size_t)layer * 4 + r) * HH;
            const float* nd = norms + 4 * (size_t)NN + (size_t)r * NN;
            float* o = nxt + (size_t)DT[r] * NH;
            if (r == 1) {  // relations 0 and 1 both feed dst ntype 0
                wmma_gemm_rowstrip<true><<<gs(strip_threads, 256), 256, 0, stream>>>(
                    agg, Wl, bl, nd, o, NN);
            } else {
                wmma_gemm_rowstrip<false><<<gs(strip_threads, 256), 256, 0, stream>>>(
                    agg, Wl, bl, nd, o, NN);
            }
        }
        relu_kernel<<<4096, 256, 0, stream>>>(nxt, 3L * NH);
        float* tmp = cur; cur = nxt; nxt = tmp;
    }

    // mean-pool per graph per ntype, summed across ntypes
    fill_zero<<<gs((long)BB * HH, 256), 256, 0, stream>>>(pooled, (long)BB * HH);
    for (int nt = 0; nt < 3; ++nt) {
        fill_zero<<<gs((long)BB * HH + BB, 256), 256, 0, stream>>>(sums, (long)BB * HH + BB);
        pool_accum<<<gs(NN, 8), 256, 0, stream>>>(cur + (size_t)nt * NH, gids[nt], sums, cnt);
        pool_combine<<<gs((long)BB * HH, 256), 256, 0, stream>>>(sums, cnt, pooled);
    }

    // projection (256x128 @ 128x512 + bias) via WMMA, then row L2-normalize
    float* out = (float*)d_out;
    const long proj_threads = (long)(BB / 16) * (512 / 16) * 32;
    wmma_gemm_tile<<<gs(proj_threads, 256), 256, 0, stream>>>(pooled, proj_W, proj_b,
                                                              out, BB, 512);
    l2norm_kernel<<<BB, 256, 0, stream>>>(out);
}